// CompanionMatrixBlock_82523501625359
// MI455X (gfx1250) — compile-verified
//
#include <hip/hip_runtime.h>
#include <hip/hip_bf16.h>

typedef float v2f __attribute__((ext_vector_type(2)));
typedef float v8f __attribute__((ext_vector_type(8)));

constexpr int kN     = 64;
constexpr int kLAGS  = 16;
constexpr int kHOR   = 128;
constexpr int kBACK  = 512;
constexpr int kBATCH = 128;
constexpr int kNL    = kN * kLAGS;   // 1024

// ---------------------------------------------------------------------------
// y_back: per (batch, 32-row time tile) workgroup, 4 waves = 4 N-tiles of 16.
// Each wave computes TWO 16x16 output tiles (time rows t0..t0+15 and
// t0+16..t0+31), reusing every loaded B fragment twice (2 independent
// accumulator chains -> better WMMA pipelining, half the W_b L2 traffic).
// wins[b,t,k] with k = l*64+j equals x[b, BACK-1-t-l, j] (0 if OOB).
// ---------------------------------------------------------------------------
__global__ __launch_bounds__(128) void yback_kernel(const float* __restrict__ x,
                                                    const float* __restrict__ Wb,
                                                    float* __restrict__ yb) {
    __shared__ __align__(16) float xLds[47][68];   // 47 rows, 64+4 pad floats

    const int b    = blockIdx.x;
    const int t0   = blockIdx.y * 32;
    const int tid  = threadIdx.x;
    const int wave = tid >> 5;        // N-tile 0..3
    const int lane = tid & 31;
    const int g    = lane >> 4;       // half-wave selector (K pair)
    const int h    = lane & 15;       // M row (A) / N col (B,D)

    // row r of xLds holds x[b, BACK-1-t0-r, :], zeros when out of range
    const int umax = kBACK - 1 - t0;
    for (int idx = tid; idx < 47 * 64; idx += 128) {
        int r = idx >> 6, j = idx & 63;
        int u = umax - r;
        xLds[r][j] = (u >= 0) ? x[((size_t)b * kBACK + u) * kN + j] : 0.0f;
    }
    __syncthreads();

    const int n0 = wave * 16;
    const float* wrow = Wb + (size_t)(n0 + h) * kNL + 2 * g;   // W_b[n][*]

    v8f c0 = {};
    v8f c1 = {};
#pragma unroll 2
    for (int l = 0; l < kLAGS; ++l) {
        const float* arow0 = &xLds[h + l][2 * g];        // tile 0: rows m+l
        const float* arow1 = &xLds[h + 16 + l][2 * g];   // tile 1: rows m+16+l
        const float* wblk  = wrow + l * 64;
#pragma unroll
        for (int cc = 0; cc < 16; ++cc) {
            v2f bf = *(const v2f*)(wblk  + 4 * cc);
            v2f a0 = *(const v2f*)(arow0 + 4 * cc);
            v2f a1 = *(const v2f*)(arow1 + 4 * cc);
            c0 = __builtin_amdgcn_wmma_f32_16x16x4_f32(false, a0, false, bf,
                                                       (short)0, c0, false, false);
            c1 = __builtin_amdgcn_wmma_f32_16x16x4_f32(false, a1, false, bf,
                                                       (short)0, c1, false, false);
        }
    }

    // D layout: c[r] -> (M = r + 8g, N = h)
    float* out = yb + ((size_t)b * kBACK + t0) * kN;
#pragma unroll
    for (int r = 0; r < 8; ++r) {
        int m = r + 8 * g;
        out[(size_t)m * kN + n0 + h]        = c0[r];
        out[(size_t)(m + 16) * kN + n0 + h] = c1[r];
    }
}

// ---------------------------------------------------------------------------
// y_horizon: 8 workgroups of 16 batches, 8 waves = 4 N-tiles x 2 K-halves.
// State (16 batches x 16 lag blocks x 64) lives in LDS as a ring buffer.
// Split-K halves the serial per-step WMMA accumulation chain (256 -> 128);
// K-half 1 dumps partials to LDS scratch, K-half 0 combines, writes the new
// ring block (slot (head+15)&15) and streams the step output to global.
// ---------------------------------------------------------------------------
__global__ __launch_bounds__(256) void yhor_kernel(const float* __restrict__ x,
                                                   const float* __restrict__ Wh,
                                                   float* __restrict__ yh) {
    __shared__ __align__(16) float sLds[16][kNL + 4];  // 16 x 1028 f32 (64.25KB)
    __shared__ __align__(16) float pScr[4][16][17];    // split-K partials

    const int b0   = blockIdx.x * 16;
    const int tid  = threadIdx.x;
    const int wave = tid >> 5;
    const int wn   = wave & 3;        // N-tile
    const int kh   = wave >> 2;       // K-half (lag blocks 0..7 / 8..15)
    const int lane = tid & 31;
    const int g    = lane >> 4;
    const int h    = lane & 15;

    // s0[b, l*64+j] = x[b, BACK-1-l, j]; head starts at 0
    for (int idx = tid; idx < 16 * kNL; idx += 256) {
        int mb = idx >> 10;
        int k  = idx & (kNL - 1);
        int l  = k >> 6, j = k & 63;
        sLds[mb][k] = x[((size_t)(b0 + mb) * kBACK + (kBACK - 1 - l)) * kN + j];
    }
    __syncthreads();

    const int n0 = wn * 16;
    const float* wrow = Wh + (size_t)(n0 + h) * kNL + 2 * g;   // W_h[n][*]

    int head = 0;
    for (int t = 0; t < kHOR; ++t) {
        v8f c = {};
#pragma unroll 2
        for (int lb = 0; lb < 8; ++lb) {
            const int l    = kh * 8 + lb;
            const int phys = (head + l) & 15;
            const float* srow = &sLds[h][phys * 64 + 2 * g];
            const float* wblk = wrow + l * 64;
#pragma unroll
            for (int cc = 0; cc < 16; ++cc) {
                v2f a  = *(const v2f*)(srow + 4 * cc);
                v2f bf = *(const v2f*)(wblk + 4 * cc);
                c = __builtin_amdgcn_wmma_f32_16x16x4_f32(false, a, false, bf,
                                                          (short)0, c, false, false);
            }
        }
        __syncthreads();                     // all state reads complete

        if (kh == 1) {                       // upper K-half: publish partials
#pragma unroll
            for (int r = 0; r < 8; ++r)
                pScr[wn][r + 8 * g][h] = c[r];
        }
        __syncthreads();

        const int hn = (head + 15) & 15;     // slot for the new block
        if (kh == 0) {                       // lower K-half: combine + commit
#pragma unroll
            for (int r = 0; r < 8; ++r) {
                int   m = r + 8 * g;
                float v = c[r] + pScr[wn][m][h];
                sLds[m][hn * 64 + n0 + h] = v;
                yh[((size_t)(b0 + m) * kHOR + t) * kN + n0 + h] = v;
            }
        }
        __syncthreads();                     // new block visible to all waves
        head = hn;
    }
}

extern "C" void kernel_launch(void* const* d_in, const int* in_sizes, int n_in,
                              void* d_out, int out_size, void* d_ws, size_t ws_size,
                              hipStream_t stream) {
    const float* x  = (const float*)d_in[0];
    const float* Wh = (const float*)d_in[1];
    const float* Wb = (const float*)d_in[2];

    float* yh = (float*)d_out;                                   // (128,128,64)
    float* yb = (float*)d_out + (size_t)kBATCH * kHOR * kN;      // (128,512,64)

    dim3 gb(kBATCH, kBACK / 32);
    yback_kernel<<<gb, 128, 0, stream>>>(x, Wb, yb);
    yhor_kernel<<<kBATCH / 16, 256, 0, stream>>>(x, Wh, yh);
}